// RainKANModel_79766132622049
// MI455X (gfx1250) — compile-verified
//
#include <hip/hip_runtime.h>
#include <hip/hip_bf16.h>

// ---------------- problem constants ----------------
#define BATCH   16384
#define HIST    192      // 24*8
#define STATD   3
#define TIMED   8
#define EMBD    16
#define IN_DIM  219      // 192+3+8+16
#define HID     64
#define NB      8        // GRID + K spline bases
#define FPI     12       // features per input, padded: [silu, 8 bases, 0,0,0]
#define K1      (IN_DIM * FPI)   // 2628, multiple of 4
#define W1ELEMS (K1 * HID)       // 168192 floats
#define W2ELEMS (HID * 9)        // 576 floats

// ---------------- tiling ----------------
#define WM             2                      // 16-row M tiles per wave
#define WAVES          8
#define BLOCK          (WAVES * 32)           // 256 threads
#define ROWS_PER_WAVE  (WM * 16)              // 32
#define ROWS_PER_BLOCK (WAVES * ROWS_PER_WAVE) // 256

typedef __attribute__((ext_vector_type(2))) float v2f;
typedef __attribute__((ext_vector_type(8))) float v8f;

// Cubic B-spline bases on uniform extended grid [-1,1], G=5, K=3 (pykan style),
// plus silu(x) in f[0]; f[9..11] are zero padding.
__device__ __forceinline__ void kan_feats(float x, float f[FPI]) {
    const float h = 0.4f;                      // 2/GRID
    float b0[11];
#pragma unroll
    for (int m = 0; m < 11; ++m) {
        float t0 = -1.0f + (float)(m - 3) * h;
        b0[m] = (x >= t0 && x < t0 + h) ? 1.0f : 0.0f;
    }
    float b1[10];
#pragma unroll
    for (int m = 0; m < 10; ++m) {
        float t0 = -1.0f + (float)(m - 3) * h;
        b1[m] = (x - t0) * 2.5f * b0[m] + ((t0 + 2.0f * h) - x) * 2.5f * b0[m + 1];
    }
    float b2[9];
#pragma unroll
    for (int m = 0; m < 9; ++m) {
        float t0 = -1.0f + (float)(m - 3) * h;
        b2[m] = (x - t0) * 1.25f * b1[m] + ((t0 + 3.0f * h) - x) * 1.25f * b1[m + 1];
    }
    const float inv3 = 1.0f / 1.2f;
    float b3[8];
#pragma unroll
    for (int m = 0; m < 8; ++m) {
        float t0 = -1.0f + (float)(m - 3) * h;
        b3[m] = (x - t0) * inv3 * b2[m] + ((t0 + 4.0f * h) - x) * inv3 * b2[m + 1];
    }
    f[0] = x / (1.0f + expf(-x));              // silu
#pragma unroll
    for (int g = 0; g < 8; ++g) f[1 + g] = b3[g];
    f[9] = 0.0f; f[10] = 0.0f; f[11] = 0.0f;
}

// Repack weights into K-major padded layouts in workspace:
//   W1p[k*64 + o], k = i*12 + j : j==0 -> w_base1[i,o]; 1..8 -> coef1[i,o,j-1]; else 0
//   W2p[j*9 ... ] per hidden unit j: [w_base2[j], coef2[j,0,0..7]]
__global__ void kan_pack_weights(const float* __restrict__ w_base1,
                                 const float* __restrict__ coef1,
                                 const float* __restrict__ w_base2,
                                 const float* __restrict__ coef2,
                                 float* __restrict__ ws) {
    int idx = blockIdx.x * blockDim.x + threadIdx.x;
    if (idx < W1ELEMS) {
        int k = idx / HID, o = idx % HID;
        int i = k / FPI, j = k % FPI;
        float v = 0.0f;
        if (j == 0)       v = w_base1[i * HID + o];
        else if (j <= NB) v = coef1[(i * HID + o) * NB + (j - 1)];
        ws[idx] = v;
    } else if (idx < W1ELEMS + W2ELEMS) {
        int e = idx - W1ELEMS;
        int i = e / 9, j = e % 9;
        ws[idx] = (j == 0) ? w_base2[i] : coef2[i * NB + (j - 1)];
    }
}

__launch_bounds__(BLOCK)
__global__ void kan_fused(const float* __restrict__ hist,
                          const float* __restrict__ stat,
                          const float* __restrict__ timef,
                          const int*   __restrict__ st_idx,
                          const float* __restrict__ emb,
                          const float* __restrict__ W1p,
                          const float* __restrict__ W2p,
                          float* __restrict__ out) {
    __shared__ float hbuf[WAVES][16][17];      // per-wave 16x16 C-tile transpose buffer

    const int tid = threadIdx.x;
    const int wave = tid >> 5;
    const int lane = tid & 31;
    const int lanelow = lane & 15;
    const int half = lane >> 4;                // 0: lanes 0-15, 1: lanes 16-31
    const int row_base = blockIdx.x * ROWS_PER_BLOCK + wave * ROWS_PER_WAVE;

    int rows[WM], stid[WM];
#pragma unroll
    for (int m = 0; m < WM; ++m) {
        rows[m] = row_base + m * 16 + lanelow; // A/C layout: lane%16 == row-in-tile
        stid[m] = st_idx[rows[m]];
    }

    v8f acc[WM][4];
#pragma unroll
    for (int m = 0; m < WM; ++m)
#pragma unroll
        for (int n = 0; n < 4; ++n)
            acc[m][n] = (v8f){0.f,0.f,0.f,0.f,0.f,0.f,0.f,0.f};

    // ---------------- Layer 1: K loop over 219 inputs, 3 WMMA K-steps each ---------
    for (int i = 0; i < IN_DIM; ++i) {
        float f[WM][FPI];
#pragma unroll
        for (int m = 0; m < WM; ++m) {
            float xv;                          // concat source select (uniform in i)
            if (i < HIST)                      xv = hist[rows[m] * HIST + i];
            else if (i < HIST + STATD)         xv = stat[rows[m] * STATD + (i - HIST)];
            else if (i < HIST + STATD + TIMED) xv = timef[rows[m] * TIMED + (i - HIST - STATD)];
            else                               xv = emb[stid[m] * EMBD + (i - HIST - STATD - TIMED)];
            kan_feats(xv, f[m]);
        }
        const int kbase = i * FPI;
#pragma unroll
        for (int t = 0; t < 3; ++t) {
            // A fragment (16x4 f32): lane holds row=lane%16, K = 2*half + {0,1}
            v2f a[WM];
#pragma unroll
            for (int m = 0; m < WM; ++m) {
                a[m].x = half ? f[m][t * 4 + 2] : f[m][t * 4 + 0];
                a[m].y = half ? f[m][t * 4 + 3] : f[m][t * 4 + 1];
            }
            const int k0 = kbase + t * 4 + half * 2;
            const float* brow0 = W1p + (size_t)k0 * HID;
            const float* brow1 = brow0 + HID;
#pragma unroll
            for (int n = 0; n < 4; ++n) {
                // B fragment (4x16 f32): lane holds col=lane%16, K = 2*half + {0,1}
                v2f b;
                b.x = brow0[n * 16 + lanelow];
                b.y = brow1[n * 16 + lanelow];
#pragma unroll
                for (int m = 0; m < WM; ++m)
                    acc[m][n] = __builtin_amdgcn_wmma_f32_16x16x4_f32(
                        false, a[m], false, b, (short)0, acc[m][n], false, false);
            }
        }
    }

    // ---------------- Layer 2: scalar KAN over h (64 -> 1), tile-by-tile ----------
#pragma unroll
    for (int m = 0; m < WM; ++m) {
        float osum = 0.0f;
        for (int ct = 0; ct < 4; ++ct) {
            __syncthreads();                   // uniform across block: buffer reuse
            // C layout: VGPR v: lanes 0-15 -> M=v, lanes 16-31 -> M=v+8, N=lane%16
#pragma unroll
            for (int v = 0; v < 8; ++v)
                hbuf[wave][v + 8 * half][lanelow] = acc[m][ct][v];
            __syncthreads();
            if (half == 0) {                   // lanes 0-15: one sample each
#pragma unroll 4
                for (int n = 0; n < 16; ++n) {
                    float hj = hbuf[wave][lanelow][n];
                    float f2[FPI];
                    kan_feats(hj, f2);
                    const int j = ct * 16 + n;
                    float s = f2[0] * W2p[j * 9];
#pragma unroll
                    for (int g = 0; g < 8; ++g)
                        s += f2[1 + g] * W2p[j * 9 + 1 + g];
                    osum += s;
                }
            }
        }
        if (half == 0)
            out[row_base + m * 16 + lanelow] = osum;
    }
}

extern "C" void kernel_launch(void* const* d_in, const int* in_sizes, int n_in,
                              void* d_out, int out_size, void* d_ws, size_t ws_size,
                              hipStream_t stream) {
    const float* hist  = (const float*)d_in[0];
    const float* stat  = (const float*)d_in[1];
    const float* timef = (const float*)d_in[2];
    const int*   stidx = (const int*)  d_in[3];
    const float* emb   = (const float*)d_in[4];
    const float* wb1   = (const float*)d_in[5];
    const float* c1    = (const float*)d_in[6];
    const float* wb2   = (const float*)d_in[7];
    const float* c2    = (const float*)d_in[8];

    float* ws  = (float*)d_ws;                 // needs ~676 KB
    float* outp = (float*)d_out;

    const int packN = W1ELEMS + W2ELEMS;
    kan_pack_weights<<<(packN + 255) / 256, 256, 0, stream>>>(wb1, c1, wb2, c2, ws);

    kan_fused<<<BATCH / ROWS_PER_BLOCK, BLOCK, 0, stream>>>(
        hist, stat, timef, stidx, emb, ws, ws + W1ELEMS, outp);
}